// stft2hcfp_79551384257069
// MI455X (gfx1250) — compile-verified
//
#include <hip/hip_runtime.h>
#include <math.h>

typedef __attribute__((ext_vector_type(2))) float v2f;
typedef __attribute__((ext_vector_type(8))) float v8f;

#define N_FFT      8000
#define NCOLS      4096          // 8 batches * 512 frames
#define NT         512
#define KBINS      352
#define HFI        2245          // HIGH_FREQ_IDX
#define HQI        583           // HIGH_QUEF_IDX
#define CEPS_ROWS  592           // 37 row-tiles * 16, rows >= 583 stored as zero
#define CUT_Q      4
#define INV_SQRT_N     0.01118033988749895f    // 1/sqrt(8000)
#define TWO_PI_OVER_N  7.853981633974483e-4f   // 2*pi/8000
#define MAP_ELEMS  (512 * 352)                 // per-batch [t][k] map
#define LDS_STRIDE 66                          // [col][n_local], keeps 8B align

__constant__ int c_hids[6] = {0, 48, 76, 96, 111, 124};

// D = A x B + C, fp32 16x16x4 WMMA
__device__ __forceinline__ v8f wmma4(v2f a, v2f b, v8f c) {
    return __builtin_amdgcn_wmma_f32_16x16x4_f32(
        false, a, false, b, (short)0, c, false, false);
}

// ---------------------------------------------------------------------------
// Kernel 1: cepstrum = nonlinear( Cos(583x8000) @ X(8000x4096) / sqrt(N) )
// Grid (128, 2): x = 32-column group, y = half of the 37 row-tiles (19 / 18).
// Each block streams dx through double-buffered LDS; next chunk's global
// loads are issued before the current chunk's 96 WMMAs (latency hidden).
// A-matrix cos values by Chebyshev recurrence c_{n+1} = 2cos(4d)c_n - c_{n-1},
// re-seeded exactly (integer phase mod 8000) every 64-deep K chunk.
// Output stored column-major: ceps_t[col * 592 + k].
// ---------------------------------------------------------------------------
__global__ __launch_bounds__(256) void k_ceps(const float* __restrict__ dx,
                                              float* __restrict__ ceps_t) {
    __shared__ float lds[2][32 * LDS_STRIDE];
    const int tid  = threadIdx.x;
    const int wave = tid >> 5;
    const int lane = tid & 31;
    const int half = lane >> 4;
    const int lid  = lane & 15;
    const int koff = 2 * half;       // K offset for this half-wave (A & B layout)
    const int c0   = blockIdx.x * 32;
    const int mbase  = blockIdx.y * 19;            // row-tile base
    const int mcount = blockIdx.y ? 18 : 19;       // tiles in this half

    v8f acc[3][2];
    #pragma unroll
    for (int mi = 0; mi < 3; ++mi)
        #pragma unroll
        for (int nt = 0; nt < 2; ++nt)
            #pragma unroll
            for (int e = 0; e < 8; ++e) acc[mi][nt][e] = 0.0f;

    // per-row-tile constants (wave-uniform activity, per-lane values)
    int   rowk_[3], pstep_[3];
    float coef_[3];
    #pragma unroll
    for (int mi = 0; mi < 3; ++mi) {
        const int ml = wave + 8 * mi;
        const int rk = (ml < mcount) ? ((mbase + ml) * 16 + lid) : 0;
        rowk_[mi]  = rk;
        pstep_[mi] = (rk * 4) % N_FFT;
        coef_[mi]  = 2.0f * __cosf((float)pstep_[mi] * TWO_PI_OVER_N);
    }

    // staging indices for the 8 cooperative loads per chunk
    int cc_[8], nl_[8];
    #pragma unroll
    for (int i = 0; i < 8; ++i) {
        const int idx = tid + i * 256;
        cc_[i] = idx >> 6;
        nl_[i] = idx & 63;
    }

    float r_[8];
    // prologue: stage chunk 0 into buffer 0
    #pragma unroll
    for (int i = 0; i < 8; ++i)
        r_[i] = dx[(size_t)(c0 + cc_[i]) * N_FFT + nl_[i]];
    #pragma unroll
    for (int i = 0; i < 8; ++i)
        lds[0][cc_[i] * LDS_STRIDE + nl_[i]] = r_[i];
    __syncthreads();

    for (int chunk = 0; chunk < 125; ++chunk) {
        const int n0c = chunk * 64;
        const float* buf = lds[chunk & 1];

        if (chunk + 1 < 125) {       // issue next chunk's global loads early
            const int n0n = n0c + 64;
            #pragma unroll
            for (int i = 0; i < 8; ++i)
                r_[i] = dx[(size_t)(c0 + cc_[i]) * N_FFT + n0n + nl_[i]];
        }

        #pragma unroll
        for (int mi = 0; mi < 3; ++mi) {
            if (wave + 8 * mi >= mcount) continue;   // wave-uniform
            const int rowk  = rowk_[mi];
            const int pstep = pstep_[mi];
            const float coef = coef_[mi];
            // exact integer phases at ks=0 and ks=-1 (mod 8000)
            int ph0 = (rowk * (n0c + koff)) % N_FFT;
            int ph1 = ph0 + rowk;  if (ph1 >= N_FFT) ph1 -= N_FFT;
            int pm0 = ph0 - pstep; if (pm0 < 0)     pm0 += N_FFT;
            int pm1 = ph1 - pstep; if (pm1 < 0)     pm1 += N_FFT;
            float c0c = __cosf((float)ph0 * TWO_PI_OVER_N);
            float c1c = __cosf((float)ph1 * TWO_PI_OVER_N);
            float c0p = __cosf((float)pm0 * TWO_PI_OVER_N);
            float c1p = __cosf((float)pm1 * TWO_PI_OVER_N);

            #pragma unroll
            for (int ks = 0; ks < 16; ++ks) {
                v2f a; a.x = c0c; a.y = c1c;
                const int fo = 4 * ks + koff;      // even -> 8B aligned LDS v2f
                const v2f b0 = *reinterpret_cast<const v2f*>(&buf[lid * LDS_STRIDE + fo]);
                const v2f b1 = *reinterpret_cast<const v2f*>(&buf[(16 + lid) * LDS_STRIDE + fo]);
                acc[mi][0] = wmma4(a, b0, acc[mi][0]);
                acc[mi][1] = wmma4(a, b1, acc[mi][1]);
                const float t0 = __builtin_fmaf(coef, c0c, -c0p);
                const float t1 = __builtin_fmaf(coef, c1c, -c1p);
                c0p = c0c; c0c = t0;
                c1p = c1c; c1c = t1;
            }
        }

        if (chunk + 1 < 125) {       // stage next chunk into the other buffer
            float* nb = lds[(chunk + 1) & 1];
            #pragma unroll
            for (int i = 0; i < 8; ++i)
                nb[cc_[i] * LDS_STRIDE + nl_[i]] = r_[i];
            __syncthreads();
        }
    }

    // epilogue: scale, power nonlinearity, zero rows k<4 and k>=583
    #pragma unroll
    for (int mi = 0; mi < 3; ++mi) {
        if (wave + 8 * mi >= mcount) continue;
        const int mt = mbase + wave + 8 * mi;
        #pragma unroll
        for (int nt = 0; nt < 2; ++nt) {
            const int col = c0 + nt * 16 + lid;    // C layout: lane = N
            float* dst = ceps_t + (size_t)col * CEPS_ROWS + mt * 16 + 8 * half;
            #pragma unroll
            for (int v = 0; v < 8; ++v) {
                const int row = mt * 16 + 8 * half + v;
                const float x = acc[mi][nt][v] * INV_SQRT_N;
                float y = 0.0f;
                if (row >= CUT_Q && row < HQI && x > 0.0f)
                    y = __expf(0.6f * __logf(x));  // x^0.6
                dst[v] = y;
            }
        }
    }
}

// ---------------------------------------------------------------------------
// Kernel 2: tfrL0[b][t][k] = sum_f W_freq[k,f] * dx[b,t,f],  f < 2245
// Grid (128, 2): x = (batch, 32-frame block), y = half of the 22 row-tiles.
// Main K loop unguarded (f+1 <= 2243), single guarded tail step.
// ---------------------------------------------------------------------------
__global__ __launch_bounds__(256) void k_tfrL0(const float* __restrict__ dx,
                                               const float* __restrict__ Wf,
                                               float* __restrict__ outT) {
    const int tid  = threadIdx.x;
    const int wave = tid >> 5;
    const int lane = tid & 31;
    const int half = lane >> 4;
    const int lid  = lane & 15;
    const int koff = 2 * half;
    const int bb = blockIdx.x >> 4;
    const int t0 = (blockIdx.x & 15) * 32;
    const int mbase = blockIdx.y * 11;             // 11 row-tiles per half

    v8f acc[2][2];
    #pragma unroll
    for (int mi = 0; mi < 2; ++mi)
        #pragma unroll
        for (int nt = 0; nt < 2; ++nt)
            #pragma unroll
            for (int e = 0; e < 8; ++e) acc[mi][nt][e] = 0.0f;

    const float* xb = dx + (size_t)bb * NT * N_FFT;
    const float* xrow0 = xb + (size_t)(t0 + lid) * N_FFT;        // nt = 0
    const float* xrow1 = xb + (size_t)(t0 + 16 + lid) * N_FFT;   // nt = 1

    for (int ks = 0; ks < 561; ++ks) {                // unguarded main loop
        const int f = 4 * ks + koff;                  // even -> 8B aligned
        const v2f bv0 = *reinterpret_cast<const v2f*>(xrow0 + f);
        const v2f bv1 = *reinterpret_cast<const v2f*>(xrow1 + f);
        #pragma unroll
        for (int mi = 0; mi < 2; ++mi) {
            if (wave + 8 * mi >= 11) continue;        // wave-uniform
            const int mt = mbase + wave + 8 * mi;
            const float* q = Wf + (size_t)(mt * 16 + lid) * HFI + f;
            v2f a; a.x = q[0]; a.y = q[1];
            acc[mi][0] = wmma4(a, bv0, acc[mi][0]);
            acc[mi][1] = wmma4(a, bv1, acc[mi][1]);
        }
    }
    {   // guarded tail: ks = 561, f = 2244 + koff
        const int f = 2244 + koff;
        v2f bv0, bv1;
        bv0.x = (f < HFI) ? xrow0[f] : 0.0f;  bv0.y = 0.0f;
        bv1.x = (f < HFI) ? xrow1[f] : 0.0f;  bv1.y = 0.0f;
        #pragma unroll
        for (int mi = 0; mi < 2; ++mi) {
            if (wave + 8 * mi >= 11) continue;
            const int mt = mbase + wave + 8 * mi;
            const float* q = Wf + (size_t)(mt * 16 + lid) * HFI + f;
            v2f a; a.x = (f < HFI) ? q[0] : 0.0f; a.y = 0.0f;
            acc[mi][0] = wmma4(a, bv0, acc[mi][0]);
            acc[mi][1] = wmma4(a, bv1, acc[mi][1]);
        }
    }

    #pragma unroll
    for (int mi = 0; mi < 2; ++mi) {
        if (wave + 8 * mi >= 11) continue;
        const int mt = mbase + wave + 8 * mi;
        #pragma unroll
        for (int nt = 0; nt < 2; ++nt) {
            const int t = t0 + nt * 16 + lid;
            float* dst = outT + (size_t)bb * MAP_ELEMS + (size_t)t * KBINS
                              + mt * 16 + 8 * half;
            #pragma unroll
            for (int v = 0; v < 8; ++v) dst[v] = acc[mi][nt][v];
        }
    }
}

// ---------------------------------------------------------------------------
// Kernel 3: tfrLQ[b][t][k] = sum_q W_quef[k,q] * ceps[q, col],  col = b*512+t
// Grid (128, 2): x = 32-column group, y = half of the 22 row-tiles.
// B operand from column-major cepstrum scratch (rows 583..591 are zero).
// Main K loop unguarded (q+1 <= 580), single guarded tail step.
// ---------------------------------------------------------------------------
__global__ __launch_bounds__(256) void k_tfrLQ(const float* __restrict__ ceps_t,
                                               const float* __restrict__ Wq,
                                               float* __restrict__ outT) {
    const int tid  = threadIdx.x;
    const int wave = tid >> 5;
    const int lane = tid & 31;
    const int half = lane >> 4;
    const int lid  = lane & 15;
    const int koff = 2 * half;
    const int c0   = blockIdx.x * 32;
    const int mbase = blockIdx.y * 11;

    v8f acc[2][2];
    #pragma unroll
    for (int mi = 0; mi < 2; ++mi)
        #pragma unroll
        for (int nt = 0; nt < 2; ++nt)
            #pragma unroll
            for (int e = 0; e < 8; ++e) acc[mi][nt][e] = 0.0f;

    const float* crow0 = ceps_t + (size_t)(c0 + lid) * CEPS_ROWS;        // nt=0
    const float* crow1 = ceps_t + (size_t)(c0 + 16 + lid) * CEPS_ROWS;   // nt=1

    for (int ks = 0; ks < 145; ++ks) {                // unguarded main loop
        const int q = 4 * ks + koff;                  // even -> 8B aligned
        const v2f bv0 = *reinterpret_cast<const v2f*>(crow0 + q);
        const v2f bv1 = *reinterpret_cast<const v2f*>(crow1 + q);
        #pragma unroll
        for (int mi = 0; mi < 2; ++mi) {
            if (wave + 8 * mi >= 11) continue;
            const int mt = mbase + wave + 8 * mi;
            const float* p = Wq + (size_t)(mt * 16 + lid) * HQI + q;
            v2f a; a.x = p[0]; a.y = p[1];
            acc[mi][0] = wmma4(a, bv0, acc[mi][0]);
            acc[mi][1] = wmma4(a, bv1, acc[mi][1]);
        }
    }
    {   // guarded tail: ks = 145, q = 580 + koff (B rows up to 583 valid/zero)
        const int q = 580 + koff;
        const v2f bv0 = *reinterpret_cast<const v2f*>(crow0 + q);
        const v2f bv1 = *reinterpret_cast<const v2f*>(crow1 + q);
        #pragma unroll
        for (int mi = 0; mi < 2; ++mi) {
            if (wave + 8 * mi >= 11) continue;
            const int mt = mbase + wave + 8 * mi;
            const float* p = Wq + (size_t)(mt * 16 + lid) * HQI + q;
            v2f a;
            a.x = (q < HQI)     ? p[0] : 0.0f;
            a.y = (q + 1 < HQI) ? p[1] : 0.0f;
            acc[mi][0] = wmma4(a, bv0, acc[mi][0]);
            acc[mi][1] = wmma4(a, bv1, acc[mi][1]);
        }
    }

    #pragma unroll
    for (int mi = 0; mi < 2; ++mi) {
        if (wave + 8 * mi >= 11) continue;
        const int mt = mbase + wave + 8 * mi;
        #pragma unroll
        for (int nt = 0; nt < 2; ++nt) {
            const int col = c0 + nt * 16 + lid;
            const int b = col >> 9, t = col & 511;
            float* dst = outT + (size_t)b * MAP_ELEMS + (size_t)t * KBINS
                              + mt * 16 + 8 * half;
            #pragma unroll
            for (int v = 0; v < 8; ++v) dst[v] = acc[mi][nt][v];
        }
    }
}

// ---------------------------------------------------------------------------
// Kernel 4: out[b][j][t][i]: j<6 -> tfrL0[b][t][i+h], j>=6 -> tfrLQ[b][t][i-h]
// One float4 (4 bins) per thread; fully coalesced writes.
// ---------------------------------------------------------------------------
__global__ __launch_bounds__(256) void k_assemble(const float* __restrict__ L0,
                                                  const float* __restrict__ LQ,
                                                  float* __restrict__ out) {
    const int g = blockIdx.x * 256 + threadIdx.x;  // (b, j, t, i4) flattened
    const int i4 = g % 88;
    int r = g / 88;
    const int t = r & 511; r >>= 9;
    const int j = r % 12;
    const int b = r / 12;
    const int h = c_hids[j % 6];
    const bool is_s = (j < 6);
    const float* src = (is_s ? L0 : LQ) + (size_t)b * MAP_ELEMS + (size_t)t * KBINS;

    float tmp[4];
    #pragma unroll
    for (int e = 0; e < 4; ++e) {
        const int i = i4 * 4 + e;
        const int s = is_s ? (i + h) : (i - h);
        tmp[e] = (s >= 0 && s < KBINS) ? src[s] : 0.0f;
    }
    *reinterpret_cast<float4*>(out + (size_t)g * 4) =
        make_float4(tmp[0], tmp[1], tmp[2], tmp[3]);
}

// ---------------------------------------------------------------------------
extern "C" void kernel_launch(void* const* d_in, const int* in_sizes, int n_in,
                              void* d_out, int out_size, void* d_ws, size_t ws_size,
                              hipStream_t stream) {
    (void)in_sizes; (void)n_in; (void)out_size; (void)ws_size;
    const float* dx = (const float*)d_in[0];
    const float* Wf = (const float*)d_in[1];
    const float* Wq = (const float*)d_in[2];
    float* out = (float*)d_out;

    float* ceps_t = (float*)d_ws;                          // 4096*592 f32
    float* L0 = ceps_t + (size_t)NCOLS * CEPS_ROWS;        // 8*512*352 f32
    float* LQ = L0 + (size_t)8 * MAP_ELEMS;                // 8*512*352 f32

    hipLaunchKernelGGL(k_ceps,     dim3(128, 2), dim3(256), 0, stream, dx, ceps_t);
    hipLaunchKernelGGL(k_tfrL0,    dim3(128, 2), dim3(256), 0, stream, dx, Wf, L0);
    hipLaunchKernelGGL(k_tfrLQ,    dim3(128, 2), dim3(256), 0, stream, ceps_t, Wq, LQ);
    hipLaunchKernelGGL(k_assemble, dim3(16896),  dim3(256), 0, stream, L0, LQ, out);
}